// MOERouter_62062277427415
// MI455X (gfx1250) — compile-verified
//
#include <hip/hip_runtime.h>
#include <hip/hip_bf16.h>

// ---------------------------------------------------------------------------
// MoE router for MI455X (gfx1250).
//  x:      [16384, 2048] f32
//  gate_w: [64, 2048]    f32
//  gate_b: [64]          f32
// outputs (flat concat in d_out):
//  logits  [16384,64] f32      @ 0
//  weights [16384,2]  f32      @ 1048576
//  indices [16384,2]  i32 bits @ 1081344
//  mask    [64,2,16384] i32    @ 1114112
// ---------------------------------------------------------------------------

#define HIDDEN    2048
#define N_EXPERTS 64
#define N_TOKENS  16384

#define OFF_W    (N_TOKENS * N_EXPERTS)          // 1048576
#define OFF_IDX  (OFF_W + N_TOKENS * 2)          // 1081344
#define OFF_MASK (OFF_IDX + N_TOKENS * 2)        // 1114112

typedef __attribute__((ext_vector_type(2))) float v2f;
typedef __attribute__((ext_vector_type(8))) float v8f;
typedef int vi4 __attribute__((__vector_size__(4 * sizeof(int))));

// --- CDNA5 async global->LDS path (guarded; falls back to sync staging) ----
#if defined(__has_builtin)
#if __has_builtin(__builtin_amdgcn_global_load_async_to_lds_b128)
#define HAVE_ASYNC_LDS 1
#endif
#endif
#ifndef HAVE_ASYNC_LDS
#define HAVE_ASYNC_LDS 0
#endif

#if HAVE_ASYNC_LDS
#if __has_builtin(__builtin_amdgcn_s_wait_asynccnt)
#define ASYNC_WAIT(n) __builtin_amdgcn_s_wait_asynccnt(n)
#else
#define ASYNC_WAIT(n) asm volatile("s_wait_asynccnt %0" ::"n"(n) : "memory")
#endif
// Builtin signature (from clang diagnostics): first arg is int4-vector* in the
// global (__device__/AS1) address space; LDS side is int4-vector* in AS3.
#define AS1P(p) ((__attribute__((address_space(1))) vi4*)(void*)(p))
#define AS3P(p) ((__attribute__((address_space(3))) vi4*)(void*)(p))
#endif

#define KC     64
#define LDSP   68                 // pitch: 272 B -> 16B-aligned rows, bank-spread
#define NCHUNK (HIDDEN / KC)      // 32

// ---------------------------------------------------------------------------
// Kernel 1: logits = x @ gate_w^T + gate_b, via V_WMMA_F32_16X16X4_F32.
// Block = 128 threads (4 waves). Each block: 64 tokens x 64 experts.
// Each wave: 16 tokens, all 64 experts (4 v8f accumulators).
// K streamed through LDS in chunks of 64; double-buffered async DMA when
// GLOBAL_LOAD_ASYNC_TO_LDS is available.
// ---------------------------------------------------------------------------
__global__ __launch_bounds__(128)
void moe_router_gemm(const float* __restrict__ x,
                     const float* __restrict__ gw,
                     const float* __restrict__ gb,
                     float* __restrict__ logits) {
#if HAVE_ASYNC_LDS
    __shared__ float xs[2][64][LDSP];   // ping/pong: 64 tokens  x KC
    __shared__ float ws[2][64][LDSP];   // ping/pong: 64 experts x KC
#else
    __shared__ float xs[1][64][LDSP];
    __shared__ float ws[1][64][LDSP];
#endif

    const int tid  = threadIdx.x;
    const int lane = tid & 31;
    const int wv   = tid >> 5;            // 0..3 -> token sub-tile
    const int tokBase = blockIdx.x * 64;

    const int m    = lane & 15;           // token row within 16 (A layout)
    const int n    = lane & 15;           // expert col within 16 (B/C layout)
    const int koff = (lane < 16) ? 0 : 2; // K pair held by this lane half

    v8f acc0 = {}, acc1 = {}, acc2 = {}, acc3 = {};

    // 16 WMMA K-steps over one staged chunk.
    auto compute_chunk = [&](const float (*xsb)[LDSP], const float (*wsb)[LDSP]) {
        #pragma unroll
        for (int kk = 0; kk < KC; kk += 4) {
            const v2f a  = *(const v2f*)&xsb[wv * 16 + m][kk + koff];
            const v2f b0 = *(const v2f*)&wsb[ 0 + n][kk + koff];
            const v2f b1 = *(const v2f*)&wsb[16 + n][kk + koff];
            const v2f b2 = *(const v2f*)&wsb[32 + n][kk + koff];
            const v2f b3 = *(const v2f*)&wsb[48 + n][kk + koff];
            acc0 = __builtin_amdgcn_wmma_f32_16x16x4_f32(false, a, false, b0, (short)0, acc0, false, false);
            acc1 = __builtin_amdgcn_wmma_f32_16x16x4_f32(false, a, false, b1, (short)0, acc1, false, false);
            acc2 = __builtin_amdgcn_wmma_f32_16x16x4_f32(false, a, false, b2, (short)0, acc2, false, false);
            acc3 = __builtin_amdgcn_wmma_f32_16x16x4_f32(false, a, false, b3, (short)0, acc3, false, false);
        }
    };

#if HAVE_ASYNC_LDS
    // Each wave DMAs its 16 rows of each tile: 8 async b128 per tile per wave
    // (one instruction moves 32 lanes x 16B = 512 B = two 64-float rows).
    const int rsub = lane >> 4;           // 0/1: which of the 2 rows
    const int acol = (lane & 15) * 4;     // 4-float column within row
    auto issue_chunk = [&](int c, int buf) {
        const int k0 = c * KC;
        #pragma unroll
        for (int j = 0; j < 8; ++j) {
            const int row = wv * 16 + j * 2 + rsub;
            __builtin_amdgcn_global_load_async_to_lds_b128(
                AS1P(x + (size_t)(tokBase + row) * HIDDEN + k0 + acol),
                AS3P(&xs[buf][row][acol]), 0, 0);
        }
        #pragma unroll
        for (int j = 0; j < 8; ++j) {
            const int row = wv * 16 + j * 2 + rsub;
            __builtin_amdgcn_global_load_async_to_lds_b128(
                AS1P(gw + (size_t)row * HIDDEN + k0 + acol),
                AS3P(&ws[buf][row][acol]), 0, 0);
        }
    };

    issue_chunk(0, 0);
    for (int c = 0; c < NCHUNK; ++c) {
        const int p = c & 1;
        if (c + 1 < NCHUNK) {
            issue_chunk(c + 1, 1 - p);   // prefetch next chunk into other buffer
            ASYNC_WAIT(16);              // chunk c's 16 DMAs (this wave) are done
        } else {
            ASYNC_WAIT(0);
        }
        __syncthreads();                 // chunk c visible from all waves
        compute_chunk(xs[p], ws[p]);
        __syncthreads();                 // done reading buf p before it is reused
    }
#else
    for (int k0 = 0; k0 < HIDDEN; k0 += KC) {
        for (int i = tid; i < 64 * (KC / 4); i += 128) {
            const int row = i >> 4;
            const int c4  = (i & 15) << 2;
            const float4 vx = *(const float4*)(x  + (size_t)(tokBase + row) * HIDDEN + k0 + c4);
            const float4 vw = *(const float4*)(gw + (size_t)row * HIDDEN + k0 + c4);
            *(float4*)&xs[0][row][c4] = vx;
            *(float4*)&ws[0][row][c4] = vw;
        }
        __syncthreads();
        compute_chunk(xs[0], ws[0]);
        __syncthreads();
    }
#endif

    // Bias, then store. C layout: lanes 0-15 VGPR r -> M=r; lanes 16-31 -> M=r+8.
    const float bb0 = gb[ 0 + n];
    const float bb1 = gb[16 + n];
    const float bb2 = gb[32 + n];
    const float bb3 = gb[48 + n];
    const int tok0  = tokBase + wv * 16;
    const int mBase = (lane < 16) ? 0 : 8;
    #pragma unroll
    for (int r = 0; r < 8; ++r) {
        float* row = logits + (size_t)(tok0 + mBase + r) * N_EXPERTS;
        row[ 0 + n] = acc0[r] + bb0;
        row[16 + n] = acc1[r] + bb1;
        row[32 + n] = acc2[r] + bb2;
        row[48 + n] = acc3[r] + bb3;
    }
}

// ---------------------------------------------------------------------------
// Kernel 2: zero the expert_mask (scatter targets written by kernel 3).
// ---------------------------------------------------------------------------
__global__ void moe_router_zero_mask(int4* __restrict__ p, int n4) {
    const int i = blockIdx.x * blockDim.x + threadIdx.x;
    if (i < n4) p[i] = make_int4(0, 0, 0, 0);
}

// ---------------------------------------------------------------------------
// Kernel 3: per-token top-2 + renormalized weights + one-hot scatter.
// softmax is monotonic: top-2 over logits == top-2 over probs, and
// w1 = p1/(p1+p2) = 1/(1 + exp(l2-l1)).
// Block stages 128 token rows through LDS with coalesced float4 loads.
// ---------------------------------------------------------------------------
__global__ __launch_bounds__(128)
void moe_router_topk(const float* __restrict__ logits,
                     float* __restrict__ wts,
                     int* __restrict__ idx,
                     int* __restrict__ mask) {
    __shared__ float lg[128][LDSP];      // 128 tokens x 64 logits (padded pitch)
    const int tid     = threadIdx.x;
    const int tokBase = blockIdx.x * 128;

    // Coalesced: 128 rows x 16 float4 = 2048 float4, 16 per thread.
    for (int i = tid; i < 128 * 16; i += 128) {
        const int row = i >> 4;
        const int c4  = (i & 15) << 2;
        *(float4*)&lg[row][c4] =
            *(const float4*)(logits + (size_t)(tokBase + row) * N_EXPERTS + c4);
    }
    __syncthreads();

    const int tok = tokBase + tid;
    float v1 = -INFINITY, v2 = -INFINITY;
    int   i1 = 0,         i2 = 0;
    #pragma unroll
    for (int e = 0; e < N_EXPERTS; e += 4) {
        const float4 q = *(const float4*)&lg[tid][e];
        const float vals[4] = { q.x, q.y, q.z, q.w };
        #pragma unroll
        for (int j = 0; j < 4; ++j) {
            const float v = vals[j];
            if (v > v1)      { v2 = v1; i2 = i1; v1 = v; i1 = e + j; }
            else if (v > v2) { v2 = v;  i2 = e + j; }
        }
    }

    const float t = expf(v2 - v1);       // <= 1, safe
    const float s = 1.0f / (1.0f + t);
    wts[tok * 2 + 0] = s;                // p1/(p1+p2)
    wts[tok * 2 + 1] = t * s;            // p2/(p1+p2)
    idx[tok * 2 + 0] = i1;
    idx[tok * 2 + 1] = i2;

    // mask[e][k][n] layout: e*(2*N_TOKENS) + k*N_TOKENS + n
    mask[(size_t)i1 * 2 * N_TOKENS + tok]            = 1;
    mask[(size_t)i2 * 2 * N_TOKENS + N_TOKENS + tok] = 1;
}

// ---------------------------------------------------------------------------
extern "C" void kernel_launch(void* const* d_in, const int* in_sizes, int n_in,
                              void* d_out, int out_size, void* d_ws, size_t ws_size,
                              hipStream_t stream) {
    const float* x  = (const float*)d_in[0];
    const float* gw = (const float*)d_in[1];
    const float* gb = (const float*)d_in[2];

    float* out    = (float*)d_out;
    float* logits = out;
    float* wts    = out + OFF_W;
    int*   idx    = (int*)(out + OFF_IDX);
    int*   mask   = (int*)(out + OFF_MASK);

    // 1) GEMM + bias -> logits
    moe_router_gemm<<<N_TOKENS / 64, 128, 0, stream>>>(x, gw, gb, logits);

    // 2) zero expert_mask (2,097,152 int32 = 524,288 int4)
    const int n4 = (N_EXPERTS * 2 * N_TOKENS) / 4;
    moe_router_zero_mask<<<(n4 + 255) / 256, 256, 0, stream>>>((int4*)mask, n4);

    // 3) top-2, weights, indices, one-hot scatter
    moe_router_topk<<<N_TOKENS / 128, 128, 0, stream>>>(logits, wts, idx, mask);
}